// CLASSIFIER_69956427317336
// MI455X (gfx1250) — compile-verified
//
#include <hip/hip_runtime.h>

typedef float v2f __attribute__((ext_vector_type(2)));
typedef float v8f __attribute__((ext_vector_type(8)));

#define FEAT 128
#define NP   256
#define NG   1024
#define NCLS 2
#define EPSV 1e-5f

// Workspace layout (in floats)
#define P0_OFF 0
#define P1_OFF (NP * FEAT)                    // 32768
#define G_OFF  (2 * NP * FEAT)                // 65536
#define A_OFF  (G_OFF + NG * FEAT)            // 196608 : A_c[p]+K_c, [p*2+c]
#define B_OFF  (A_OFF + NP * NCLS)            // 197120 : B_c[g],     [g*2+c]
// total floats = B_OFF + NG*NCLS = 199168  (~797 KB)

// ---------------------------------------------------------------------------
// Kernel 1: build transformed operands + per-row scalars.
//   rows [0, NP)        : probe rows  -> P0, P1, A
//   rows [NP, NP+NG)    : gallery rows-> G, B
// One block per row, 128 threads (one per feature), LDS tree reduction.
// ---------------------------------------------------------------------------
__global__ __launch_bounds__(FEAT)
void classifier_prep(const float* __restrict__ probe,
                     const float* __restrict__ gallery,
                     const float* __restrict__ bnw,
                     const float* __restrict__ bnb,
                     const float* __restrict__ bnm,
                     const float* __restrict__ bnv,
                     const float* __restrict__ W,
                     const float* __restrict__ lb,
                     float* __restrict__ ws) {
    const int row = blockIdx.x;
    const int f   = threadIdx.x;

    __shared__ float s0[FEAT];
    __shared__ float s1[FEAT];

    const float inv = bnw[f] * rsqrtf(bnv[f] + EPSV);
    const float w0  = W[f] * inv;            // Ws[0,f]
    const float w1  = W[FEAT + f] * inv;     // Ws[1,f]

    float* P0 = ws + P0_OFF;
    float* P1 = ws + P1_OFF;
    float* G  = ws + G_OFF;
    float* A  = ws + A_OFF;
    float* B  = ws + B_OFF;

    if (row < NP) {
        const float x = probe[row * FEAT + f];
        P0[row * FEAT + f] = x * w0;
        P1[row * FEAT + f] = x * w1;
        // BN/bias constant term K_c, distributed over features; linear bias on f==0
        const float kc = bnb[f] - bnm[f] * inv;
        float t0 = w0 * x * x + W[f] * kc;
        float t1 = w1 * x * x + W[FEAT + f] * kc;
        if (f == 0) { t0 += lb[0]; t1 += lb[1]; }
        s0[f] = t0;
        s1[f] = t1;
        __syncthreads();
        #pragma unroll
        for (int s = FEAT / 2; s > 0; s >>= 1) {
            if (f < s) { s0[f] += s0[f + s]; s1[f] += s1[f + s]; }
            __syncthreads();
        }
        if (f == 0) {
            A[row * NCLS + 0] = s0[0];
            A[row * NCLS + 1] = s1[0];
        }
    } else {
        const int g = row - NP;
        const float y = gallery[g * FEAT + f];
        G[g * FEAT + f] = -2.0f * y;
        s0[f] = w0 * y * y;
        s1[f] = w1 * y * y;
        __syncthreads();
        #pragma unroll
        for (int s = FEAT / 2; s > 0; s >>= 1) {
            if (f < s) { s0[f] += s0[f + s]; s1[f] += s1[f + s]; }
            __syncthreads();
        }
        if (f == 0) {
            B[g * NCLS + 0] = s0[0];
            B[g * NCLS + 1] = s1[0];
        }
    }
}

// ---------------------------------------------------------------------------
// Kernel 2: two K=128 f32 WMMA GEMMs with shared B operand + rank-1 epilogue.
// Block = 256 threads = 8 wave32s. Each wave owns a 16(M probe) x 16(N gallery)
// tile for BOTH classes; a block covers 16 probes x 128 galleries.
// Grid = (NG/128, NP/16) = (8, 16).
//
// V_WMMA_F32_16X16X4_F32 operand layout (ISA 7.12.2):
//   A 16x4 : lane l holds row M=(l&15), K = 2*(l>>4) + {0,1}  -> float2 load
//   B 4x16 : lane l holds col N=(l&15), K = 2*(l>>4) + {0,1}  -> float2 load
//   C/D    : VGPR i = row M = i + 8*(l>>4), col N = (l&15)
// ---------------------------------------------------------------------------
__global__ __launch_bounds__(256)
void classifier_wmma(const float* __restrict__ ws,
                     float* __restrict__ out) {
    const float* P0 = ws + P0_OFF;
    const float* P1 = ws + P1_OFF;
    const float* G  = ws + G_OFF;
    const float* A  = ws + A_OFF;
    const float* B  = ws + B_OFF;

    const int lane = threadIdx.x & 31;
    const int wave = threadIdx.x >> 5;
    const int rlo  = lane & 15;
    const int hi   = lane >> 4;

    const int m0 = blockIdx.y * 16;               // probe tile base
    const int n0 = blockIdx.x * 128 + wave * 16;  // gallery tile base

    const float* pa0 = P0 + (m0 + rlo) * FEAT + 2 * hi;
    const float* pa1 = P1 + (m0 + rlo) * FEAT + 2 * hi;
    const float* pb  = G  + (n0 + rlo) * FEAT + 2 * hi;

    v8f c0 = {};
    v8f c1 = {};

    #pragma unroll
    for (int k = 0; k < FEAT; k += 4) {
        const v2f a0 = *(const v2f*)(pa0 + k);
        const v2f a1 = *(const v2f*)(pa1 + k);
        const v2f bf = *(const v2f*)(pb  + k);
        c0 = __builtin_amdgcn_wmma_f32_16x16x4_f32(
                 false, a0, false, bf, (short)0, c0, false, false);
        c1 = __builtin_amdgcn_wmma_f32_16x16x4_f32(
                 false, a1, false, bf, (short)0, c1, false, false);
    }

    // Epilogue: out[p,g,c] = cross_c + A_c[p] + B_c[g]; classes packed -> float2
    const int g   = n0 + rlo;
    const float bg0 = B[g * NCLS + 0];
    const float bg1 = B[g * NCLS + 1];

    #pragma unroll
    for (int i = 0; i < 8; ++i) {
        const int p = m0 + i + 8 * hi;
        v2f o;
        o.x = c0[i] + A[p * NCLS + 0] + bg0;
        o.y = c1[i] + A[p * NCLS + 1] + bg1;
        *(v2f*)(out + ((size_t)p * NG + g) * NCLS) = o;
    }
}

// ---------------------------------------------------------------------------
extern "C" void kernel_launch(void* const* d_in, const int* in_sizes, int n_in,
                              void* d_out, int out_size, void* d_ws, size_t ws_size,
                              hipStream_t stream) {
    const float* probe   = (const float*)d_in[0];
    const float* gallery = (const float*)d_in[1];
    const float* bnw     = (const float*)d_in[2];
    const float* bnb     = (const float*)d_in[3];
    const float* bnm     = (const float*)d_in[4];
    const float* bnv     = (const float*)d_in[5];
    const float* W       = (const float*)d_in[6];
    const float* lb      = (const float*)d_in[7];
    float* out = (float*)d_out;
    float* ws  = (float*)d_ws;

    classifier_prep<<<NP + NG, FEAT, 0, stream>>>(
        probe, gallery, bnw, bnb, bnm, bnv, W, lb, ws);

    dim3 grid(NG / 128, NP / 16);
    classifier_wmma<<<grid, 256, 0, stream>>>(ws, out);
}